// F2FPoseModel_23605140259574
// MI455X (gfx1250) — compile-verified
//
#include <hip/hip_runtime.h>
#include <stdint.h>

#define NPTS 5760
#define NB   4
#define TK   64              // columns staged per LDS tile
#define LSTRIDE 68           // padded data-tile row stride in dwords (16B rows, conflict-free b64)
#define TGSTRIDE 68          // padded tgt-tile row stride in dwords
#define TGTOFF (16 * LSTRIDE)                    // float offset of tgt tile inside buffer
#define BUFFLOATS (16 * LSTRIDE + 4 * TGSTRIDE)  // 1088 + 272 = 1360 floats per buffer
#define NTILES (NPTS / TK)   // 90
#define MASKEDV (-3.0e38f)   // finite sentinel: all-masked rows -> uniform softmax (matches reference)
#define NEGV    (-1.0e30f)

typedef __attribute__((ext_vector_type(2))) float v2f;
typedef __attribute__((ext_vector_type(8))) float v8f;

// ---- CDNA5 async global->LDS (ASYNCcnt path) -------------------------------
__device__ __forceinline__ unsigned ldsOffset(const void* p) {
  // flat LDS pointer: low 32 bits are the LDS byte offset (ISA 10.2 aperture rules)
  return (unsigned)(unsigned long long)p;
}
__device__ __forceinline__ void asyncLoadB128(unsigned ldsOff, unsigned gOff, const void* saddr) {
  asm volatile("global_load_async_to_lds_b128 %0, %1, %2"
               :: "v"(ldsOff), "v"(gOff), "s"(saddr) : "memory");
}
__device__ __forceinline__ void waitAsyncLE10() { asm volatile("s_wait_asynccnt 10" ::: "memory"); }
__device__ __forceinline__ void waitAsync0()   { asm volatile("s_wait_asynccnt 0"  ::: "memory"); }

// ---- kernel 0: init workspace + output -------------------------------------
__global__ void k_init(unsigned long long* __restrict__ colkeys, float* __restrict__ out) {
  int i = blockIdx.x * 256 + threadIdx.x;
  if (i < NB * NPTS) colkeys[i] = 0ull;
  if (i == 0) out[0] = 0.0f;
}

// ---- kernel 1: fused online-softmax row pass with WMMA ---------------------
// one wave handles 16 rows; block = 4 waves; grid = (NPTS/64, NB)
__global__ __launch_bounds__(128)
void k_rowpass(const float* __restrict__ mv, const float* __restrict__ tgt,
               const int* __restrict__ pmask,
               float* __restrict__ p2u, float* __restrict__ rsum,
               int* __restrict__ i2to1) {
  __shared__ unsigned s_m2b[NPTS / 32];                    // packed m2 bitmask (180 words)
  __shared__ __align__(16) float s_tile[4][2][BUFFLOATS];  // per-wave double buffers

  const int tid  = threadIdx.x;
  const int lane = tid & 31;
  const int w    = tid >> 5;
  const int b    = blockIdx.y;
  const int rowbase = (blockIdx.x * 4 + w) * 16;

  const float* mvb  = mv  + (size_t)b * NPTS * NPTS;
  const float* tgtb = tgt + (size_t)b * 3 * NPTS;
  const int*   m2   = pmask + (size_t)(2 * b + 1) * NPTS;

  for (int j = tid; j < NPTS / 32; j += 128) {
    unsigned wd = 0;
    const int* mp = m2 + j * 32;
#pragma unroll
    for (int k = 0; k < 32; ++k) wd |= (mp[k] != 0 ? 1u : 0u) << k;
    s_m2b[j] = wd;
  }
  __syncthreads();

  const int r     = lane & 15;   // softmax row within group (and B column index n)
  const int khalf = lane >> 4;   // 0 -> K={0,1}, 1 -> K={2,3}
  const int c16   = lane & 15;   // 16-byte chunk id for staging
  const int trow  = (r < 3) ? r : 3;  // clamped tgt-tile row (row 3 = staged dup of z)

  v8f acc = {};                  // unnormalized points2 accumulator (C matrix)
  float m = MASKEDV;             // running row max (scaled domain)
  float s = 0.0f;                // running exp-sum
  float bestv = -INFINITY;       // row argmax tracking (raw masked values)
  int   besti = 0;

  // stage tile tt into a buffer: 8 data b128 + 2 tgt b128 (10 async ops)
#define STAGE_TILE(LBASE, TT)                                                        \
  do {                                                                               \
    _Pragma("unroll")                                                                \
    for (int i = 0; i < 8; ++i) {                                                    \
      int row = 2 * i + khalf;                                                       \
      unsigned loff = (LBASE) + (unsigned)(row * LSTRIDE + c16 * 4) * 4u;            \
      unsigned goff = (unsigned)((rowbase + row) * NPTS + (TT) * TK + c16 * 4) * 4u; \
      asyncLoadB128(loff, goff, mvb);                                                \
    }                                                                                \
    { /* tgt rows x (lanes<16) and y (lanes>=16) */                                  \
      unsigned loff = (LBASE) + (unsigned)(TGTOFF + khalf * TGSTRIDE + c16 * 4) * 4u;\
      unsigned goff = (unsigned)(khalf * NPTS + (TT) * TK + c16 * 4) * 4u;           \
      asyncLoadB128(loff, goff, tgtb);                                               \
    }                                                                                \
    { /* tgt row z into rows 2 and 3 (dup keeps clamped reads valid) */              \
      unsigned loff = (LBASE) +                                                      \
          (unsigned)(TGTOFF + (2 + khalf) * TGSTRIDE + c16 * 4) * 4u;                \
      unsigned goff = (unsigned)(2 * NPTS + (TT) * TK + c16 * 4) * 4u;               \
      asyncLoadB128(loff, goff, tgtb);                                               \
    }                                                                                \
  } while (0)

  // prologue: stage tile 0 into buffer 0
  {
    unsigned lbase0 = ldsOffset(&s_tile[w][0][0]);
    STAGE_TILE(lbase0, 0);
  }

  for (int tt = 0; tt < NTILES; ++tt) {
    const int buf = tt & 1;
    if (tt + 1 < NTILES) {
      unsigned lbase = ldsOffset(&s_tile[w][buf ^ 1][0]);
      STAGE_TILE(lbase, tt + 1);
      waitAsyncLE10();   // previous buffer's 10 async ops complete (in-order done)
    } else {
      waitAsync0();
    }

    const float* tile = &s_tile[w][buf][0];
    const float* tgtt = tile + TGTOFF;
    const unsigned mw0 = s_m2b[tt * 2];       // mask bits for cols [tt*64, +32)
    const unsigned mw1 = s_m2b[tt * 2 + 1];   // mask bits for cols [tt*64+32, +32)

#pragma unroll 8
    for (int sub = 0; sub < 16; ++sub) {
      const int cl = sub * 4 + khalf * 2;     // local col of this lane's K pair
      const int cg = tt * TK + cl;            // global col
      float d0 = tile[r * LSTRIDE + cl];
      float d1 = tile[r * LSTRIDE + cl + 1];
      const unsigned mw = (sub < 8) ? mw0 : mw1;
      const int bit = cl & 31;                // even, so bit+1 stays in word
      const bool mm0 = (mw >> bit) & 1u;
      const bool mm1 = (mw >> (bit + 1)) & 1u;

      // row argmax on raw masked values (first-occurrence on ties via strict >)
      float raw0 = mm0 ? d0 : NEGV;
      float raw1 = mm1 ? d1 : NEGV;
      if (raw0 > bestv) { bestv = raw0; besti = cg; }
      if (raw1 > bestv) { bestv = raw1; besti = cg + 1; }

      // scaled values for softmax (TEMP = 100)
      float sv0 = mm0 ? d0 * 100.0f : MASKEDV;
      float sv1 = mm1 ? d1 * 100.0f : MASKEDV;

      float tl = fmaxf(sv0, sv1);
      float tr = fmaxf(tl, __shfl_xor(tl, 16));
      float mn = fmaxf(m, tr);
      if (__any(mn > m)) {                     // wave-uniform, rare rescale event
        float fac = __expf(m - mn);            // ==1 for rows with no update
        s *= fac;
#pragma unroll
        for (int k = 0; k < 8; ++k) {
          float f = __shfl(fac, (lane & 16) ? (k + 8) : k);  // factor for C row
          acc[k] *= f;
        }
      }
      m = mn;

      float p0 = __expf(sv0 - m);
      float p1 = __expf(sv1 - m);
      s += p0 + p1;

      // B matrix: 4x16, cols 0..2 = tgt x/y/z at points cg..; rest 0.
      // Unconditional LDS read from clamped row + cndmask (no EXEC divergence).
      float tv0 = tgtt[trow * TGSTRIDE + cl];
      float tv1 = tgtt[trow * TGSTRIDE + cl + 1];
      float b0 = (r < 3) ? tv0 : 0.0f;
      float b1 = (r < 3) ? tv1 : 0.0f;

      v2f a;  a.x = p0; a.y = p1;
      v2f bb; bb.x = b0; bb.y = b1;
      acc = __builtin_amdgcn_wmma_f32_16x16x4_f32(false, a, false, bb,
                                                  (short)0, acc, false, false);
    }
  }

  // combine lane-pair state (rows are duplicated in lanes l and l+16)
  float stot = s + __shfl_xor(s, 16);
  float bv2  = __shfl_xor(bestv, 16);
  int   bi2  = __shfl_xor(besti, 16);
  if (bv2 > bestv || (bv2 == bestv && bi2 < besti)) { bestv = bv2; besti = bi2; }
  if (lane < 16) {
    rsum [(size_t)b * NPTS + rowbase + lane] = stot;
    i2to1[(size_t)b * NPTS + rowbase + lane] = besti;
  }

  // write unnormalized points2 from the C layout (VGPR k -> rows k / k+8)
  const int ccol = lane & 15;
  const int rtop = khalf * 8;
  if (ccol < 3) {
#pragma unroll
    for (int k = 0; k < 8; ++k) {
      p2u[((size_t)b * NPTS + rowbase + rtop + k) * 3 + ccol] = acc[k];
    }
  }
}

// ---- kernel 2: column argmax (ind1to2) -------------------------------------
__global__ __launch_bounds__(256)
void k_colpass(const float* __restrict__ mv, const int* __restrict__ pmask,
               unsigned long long* __restrict__ colkeys) {
  const int b = blockIdx.z;
  const int j = blockIdx.x * 256 + threadIdx.x;
  if (j >= NPTS) return;
  const int rows = NPTS / 8;
  const int r0 = blockIdx.y * rows, r1 = r0 + rows;
  const float* mvb = mv + (size_t)b * NPTS * NPTS;
  const int* m1 = pmask + (size_t)(2 * b) * NPTS;

  float best = -INFINITY; int bidx = 0;
  for (int i0 = r0; i0 < r1; i0 += 4) {
#pragma unroll
    for (int u = 0; u < 4; ++u) {
      int i = i0 + u;
      float v = mvb[(size_t)i * NPTS + j];
      v = m1[i] ? v : NEGV;
      if (v > best) { best = v; bidx = i; }
    }
  }
  unsigned ub = __float_as_uint(best);
  ub = (ub & 0x80000000u) ? ~ub : (ub | 0x80000000u);   // order-preserving map
  unsigned long long key = ((unsigned long long)ub << 32) | (unsigned)(~(unsigned)bidx);
  atomicMax(&colkeys[(size_t)b * NPTS + j], key);       // ties -> smallest index
}

// ---- kernel 3: per-point loss + reduction ----------------------------------
__global__ __launch_bounds__(256)
void k_finalize(const float* __restrict__ src, const float* __restrict__ wts,
                const float* __restrict__ Tiv, const int* __restrict__ pmask,
                const float* __restrict__ p2u, const float* __restrict__ rsum,
                const int* __restrict__ i2to1,
                const unsigned long long* __restrict__ colkeys,
                float* __restrict__ out) {
  const int b = blockIdx.x;
  const int tid = threadIdx.x;
  const float* Ts = Tiv + (size_t)(2 * b) * 16;
  const float* Tt = Tiv + (size_t)(2 * b + 1) * 16;

  // A = se3_inv(T_tgt) rows 0..2 ; T21 = A @ T_src (T_src last row is general)
  float A[3][4], T21[3][4];
#pragma unroll
  for (int i = 0; i < 3; ++i) {
#pragma unroll
    for (int j = 0; j < 3; ++j) A[i][j] = Tt[j * 4 + i];
    A[i][3] = -(Tt[0 * 4 + i] * Tt[3] + Tt[1 * 4 + i] * Tt[7] + Tt[2 * 4 + i] * Tt[11]);
  }
#pragma unroll
  for (int i = 0; i < 3; ++i)
#pragma unroll
    for (int j = 0; j < 4; ++j)
      T21[i][j] = A[i][0] * Ts[j] + A[i][1] * Ts[4 + j] + A[i][2] * Ts[8 + j] + A[i][3] * Ts[12 + j];

  const float* srcb = src + (size_t)b * 3 * NPTS;
  const float* wb   = wts + (size_t)b * 6 * NPTS;
  const int*   m1   = pmask + (size_t)(2 * b) * NPTS;

  float cnt = 0.0f, smah = 0.0f, sd = 0.0f;
  for (int i = tid; i < NPTS; i += 256) {
    float px = srcb[i], py = srcb[NPTS + i], pz = srcb[2 * NPTS + i];
    float inv = 1.0f / rsum[(size_t)b * NPTS + i];
    size_t pb = ((size_t)b * NPTS + i) * 3;
    float e0 = T21[0][0] * px + T21[0][1] * py + T21[0][2] * pz + T21[0][3] - p2u[pb + 0] * inv;
    float e1 = T21[1][0] * px + T21[1][1] * py + T21[1][2] * pz + T21[1][3] - p2u[pb + 1] * inv;
    float e2 = T21[2][0] * px + T21[2][1] * py + T21[2][2] * pz + T21[2][3] - p2u[pb + 2] * inv;
    float w0 = wb[i], w1 = wb[NPTS + i], w2 = wb[2 * NPTS + i];
    float d0 = wb[3 * NPTS + i], d1 = wb[4 * NPTS + i], d2 = wb[5 * NPTS + i];
    float u0 = e0 + w0 * e1 + w1 * e2;
    float u1 = e1 + w2 * e2;
    float u2 = e2;
    float mah = __expf(d0) * u0 * u0 + __expf(d1) * u1 * u1 + __expf(d2) * u2 * u2;
    int i2 = i2to1[(size_t)b * NPTS + i];
    unsigned back = ~(unsigned)(colkeys[(size_t)b * NPTS + i2] & 0xFFFFFFFFull);
    bool inl = (m1[i] != 0) && ((int)back == i) && (mah < 10000.0f);
    if (inl) { cnt += 1.0f; smah += mah; sd += d0 + d1 + d2; }
  }

  __shared__ float red[3][256];
  red[0][tid] = cnt; red[1][tid] = smah; red[2][tid] = sd;
  __syncthreads();
  for (int off = 128; off > 0; off >>= 1) {
    if (tid < off) {
      red[0][tid] += red[0][tid + off];
      red[1][tid] += red[1][tid + off];
      red[2][tid] += red[2][tid + off];
    }
    __syncthreads();
  }
  if (tid == 0) {
    float c = fmaxf(red[0][0], 1.0f);
    atomicAdd(out, red[1][0] / c - red[2][0] / c);
  }
}

// ---- host launch -----------------------------------------------------------
extern "C" void kernel_launch(void* const* d_in, const int* in_sizes, int n_in,
                              void* d_out, int out_size, void* d_ws, size_t ws_size,
                              hipStream_t stream) {
  const float* src = (const float*)d_in[0];   // (4,3,5760)
  const float* tgt = (const float*)d_in[1];   // (4,3,5760)
  const float* wts = (const float*)d_in[2];   // (4,6,5760)
  const float* mv  = (const float*)d_in[3];   // (4,5760,5760)
  const float* Tiv = (const float*)d_in[4];   // (8,4,4)
  const int*   pm  = (const int*)d_in[5];     // (8,5760)
  float* out = (float*)d_out;

  char* ws = (char*)d_ws;
  unsigned long long* colkeys = (unsigned long long*)ws;          // 4*5760*8
  float* p2u   = (float*)(ws + (size_t)NB * NPTS * 8);            // 4*5760*3*4
  float* rsum  = (float*)((char*)p2u + (size_t)NB * NPTS * 3 * 4);
  int*   i2to1 = (int*)((char*)rsum + (size_t)NB * NPTS * 4);

  k_init<<<(NB * NPTS + 255) / 256, 256, 0, stream>>>(colkeys, out);
  k_rowpass<<<dim3(NPTS / TK, NB), 128, 0, stream>>>(mv, tgt, pm, p2u, rsum, i2to1);
  k_colpass<<<dim3((NPTS + 255) / 256, 8, NB), 256, 0, stream>>>(mv, pm, colkeys);
  k_finalize<<<NB, 256, 0, stream>>>(src, wts, Tiv, pm, p2u, rsum, i2to1, colkeys, out);
}